// NeuralDWAV_9491877724763
// MI455X (gfx1250) — compile-verified
//
#include <hip/hip_runtime.h>
#include <cmath>
#include <stdint.h>

typedef __attribute__((ext_vector_type(2))) float v2f;
typedef __attribute__((ext_vector_type(8))) float v8f;

// db4 analysis filters (compile-time; identical to the reference's constants)
__constant__ float c_lp[8] = {
  -0.010597401784997278f,  0.032883011666982945f,  0.030841381835986965f,
  -0.18703481171888114f,  -0.02798376941698385f,   0.6308807679295904f,
   0.7148465705525415f,    0.23037781330885523f };
// hp[t] = lo[7-t] * (-1)^t
__constant__ float c_hp[8] = {
   0.23037781330885523f,  -0.7148465705525415f,    0.6308807679295904f,
   0.02798376941698385f,  -0.18703481171888114f,  -0.030841381835986965f,
   0.032883011666982945f,  0.010597401784997278f };

__device__ __forceinline__ float tapf(const float* f, int t) {
  return (t >= 0 && t < 8) ? f[t] : 0.0f;
}

// y * (sigmoid(10(y-b)) + sigmoid(-10(y+b)))
__device__ __forceinline__ float gate(float y, float b) {
  float s1 = 1.0f / (1.0f + __expf(-10.0f * (y - b)));
  float s2 = 1.0f / (1.0f + __expf( 10.0f * (y + b)));
  return y * (s1 + s2);
}

// async DMA of one dword global -> LDS (ASYNCcnt-tracked, bypasses VGPRs)
__device__ __forceinline__ void async_g2l_b32(unsigned lds_off, const void* gaddr) {
  asm volatile("global_load_async_to_lds_b32 %0, %1, off"
               :: "v"(lds_off), "v"((unsigned long long)(uintptr_t)gaddr)
               : "memory");
}
__device__ __forceinline__ void wait_async0() {
  asm volatile("s_wait_asynccnt 0x0" ::: "memory");
}

// ---------------------------------------------------------------------------
// Analysis: (B,Gin,Lin) -> (B,2*Gin,Lin/2), stride-2 8-tap conv + gate.
// Block: 16 batches x 1 group x 128 output positions; 8 waves, 1 WMMA tile each.
// WMMA tile: D[16 batch x 16 pos] = sum_k A_data(16x4) * B_band(4x16).
//   (data as A, filter band matrix as B -> lanes = output positions ->
//    fully coalesced 64B result stores)
// ---------------------------------------------------------------------------
#define AN_SPAN   264     // input samples staged per row (2*128 + 8)
#define AN_STRIDE 265     // odd stride -> conflict-free 16-lane column reads

__global__ __launch_bounds__(256)
void wpt_analysis_wmma(const float* __restrict__ in, float* __restrict__ out,
                       const float* __restrict__ bias,
                       int Gin, int Lin) {
  const int Lout = Lin >> 1;
  const int Gout = Gin << 1;
  const int g    = blockIdx.z;
  const int bb0  = blockIdx.y * 16;
  const int M0   = blockIdx.x * 128;   // first output sample of this block
  const int s0   = 2 * M0 - 3;         // first input sample needed (pad = 3)

  __shared__ float lds[16 * AN_STRIDE];

  const int tid = threadIdx.x;
  // async DMA staging (coalesced per row); ds_store zeros for the pad edges
  for (int idx = tid; idx < 16 * AN_SPAN; idx += 256) {
    int n = idx / AN_SPAN, col = idx - n * AN_SPAN;
    int s = s0 + col;
    float* dst = &lds[n * AN_STRIDE + col];
    if (s >= 0 && s < Lin) {
      const float* gp = in + ((size_t)(bb0 + n) * Gin + g) * (size_t)Lin + s;
      async_g2l_b32((unsigned)(uintptr_t)dst, gp);
    } else {
      *dst = 0.0f;
    }
  }
  wait_async0();
  __syncthreads();

  const int wave = tid >> 5;
  const int lane = tid & 31;
  const int half = lane >> 4;    // 0: lanes 0-15, 1: lanes 16-31
  const int n    = lane & 15;    // A row (batch) == D column (position) index

  // Constant band-matrix B fragments: B[o][m] = h[o - 2m], o = 4k + 2*half (+1)
  v2f F_lp[10], F_hp[10];
#pragma unroll
  for (int k = 0; k < 10; ++k) {
    int t = 4 * k + 2 * half - 2 * n;
    F_lp[k].x = tapf(c_lp, t);  F_lp[k].y = tapf(c_lp, t + 1);
    F_hp[k].x = tapf(c_hp, t);  F_hp[k].y = tapf(c_hp, t + 1);
  }

  v8f acc_lp = {}; v8f acc_hp = {};
  const int colbase = 32 * wave;
#pragma unroll
  for (int k = 0; k < 10; ++k) {
    int c = colbase + 4 * k + 2 * half;
    v2f dat;                                  // A fragment: row = batch n
    dat.x = lds[n * AN_STRIDE + c];
    dat.y = lds[n * AN_STRIDE + c + 1];
    acc_lp = __builtin_amdgcn_wmma_f32_16x16x4_f32(false, dat, false, F_lp[k],
                                                   (short)0, acc_lp, false, false);
    acc_hp = __builtin_amdgcn_wmma_f32_16x16x4_f32(false, dat, false, F_hp[k],
                                                   (short)0, acc_hp, false, false);
  }

  // channel routing: group j even -> rows [lp,hp], odd -> [hp,lp]
  const int ch_lp = 2 * g + (g & 1);
  const int ch_hp = 2 * g + 1 - (g & 1);
  const float bl = bias[ch_lp];
  const float bh = bias[ch_hp];
  const int m = M0 + 16 * wave + n;           // lane's output sample

#pragma unroll
  for (int d = 0; d < 8; ++d) {
    int bat = bb0 + d + 8 * half;             // C/D layout: rows d and d+8
    size_t rowo = (size_t)bat * Gout;
    out[(rowo + ch_lp) * (size_t)Lout + m] = gate(acc_lp[d], bl);
    out[(rowo + ch_hp) * (size_t)Lout + m] = gate(acc_hp[d], bh);
  }
}

// ---------------------------------------------------------------------------
// Synthesis: (B,2G,M) -> (B,G,2M). Transposed conv with lhs_dilation=2,
// pad (7,7), then slice [3:3+2M]. With interleaved input z[2m+c]=y[c,m] this
// is a band matmul with parity-alternating rows, span 24 -> 6 WMMA per tile.
// ---------------------------------------------------------------------------
#define SY_MROWS 68
#define SY_MSTR  69
#define SY_ROWSTR (2 * SY_MSTR)   // 138 -> conflict-free column reads

__global__ __launch_bounds__(256)
void wpt_synthesis_wmma(const float* __restrict__ in, float* __restrict__ out,
                        int G, int M) {
  const int L2  = 2 * M;
  const int g   = blockIdx.z;
  const int bb0 = blockIdx.y * 16;
  const int Q0  = blockIdx.x * 128;     // first output sample q of this block
  const int Mbase = (Q0 - 4) >> 1;      // first staged m index (may be -2)
  const int Gin = 2 * G;

  __shared__ float lds[16 * SY_ROWSTR];

  const int tid = threadIdx.x;
  for (int idx = tid; idx < 16 * 2 * SY_MROWS; idx += 256) {
    int n  = idx / (2 * SY_MROWS);
    int rem = idx - n * (2 * SY_MROWS);
    int c  = rem / SY_MROWS, ml = rem - c * SY_MROWS;
    int m  = Mbase + ml;
    float* dst = &lds[n * SY_ROWSTR + c * SY_MSTR + ml];
    if (m >= 0 && m < M) {
      const float* gp = in + ((size_t)(bb0 + n) * Gin + 2 * g + c) * (size_t)M + m;
      async_g2l_b32((unsigned)(uintptr_t)dst, gp);
    } else {
      *dst = 0.0f;
    }
  }
  wait_async0();
  __syncthreads();

  const int wave = tid >> 5;
  const int lane = tid & 31;
  const int half = lane >> 4;
  const int n = lane & 15;

  // K rows for this group: [lp,hp] if g even else [hp,lp]
  const float* k0 = (g & 1) ? c_hp : c_lp;   // channel c=0 filter row
  const float* k1 = (g & 1) ? c_lp : c_hp;   // channel c=1 filter row

  // B fragments: band matrix, rows K=o, cols N=q-position r=n
  v2f F[6];
#pragma unroll
  for (int k = 0; k < 6; ++k) {
    float vv[2];
#pragma unroll
    for (int e = 0; e < 2; ++e) {
      int oo = 4 * k + 2 * half + e;
      int d  = ((n & 1) == 0) ? (oo - n) : (oo - n - 1);
      float v = 0.0f;
      if (d >= 0 && d < 8) {
        const float* kr = (d & 1) ? k1 : k0;
        int u = d >> 1;
        v = ((n & 1) == 0) ? kr[7 - 2 * u] : kr[6 - 2 * u];
      }
      vv[e] = v;
    }
    F[k].x = vv[0]; F[k].y = vv[1];
  }

  v8f acc = {};
  const int mb = 8 * wave;               // (16*wave)/2
#pragma unroll
  for (int k = 0; k < 6; ++k) {
    int ml = mb + 2 * k + half;          // (4k + 2*half)/2
    v2f dat;                             // A fragment: row = batch n
    dat.x = lds[n * SY_ROWSTR + ml];               // c = 0 (p even)
    dat.y = lds[n * SY_ROWSTR + SY_MSTR + ml];     // c = 1 (p odd, same m)
    acc = __builtin_amdgcn_wmma_f32_16x16x4_f32(false, dat, false, F[k],
                                                (short)0, acc, false, false);
  }

  const int q = Q0 + 16 * wave + n;      // lane's output sample -> coalesced
#pragma unroll
  for (int d = 0; d < 8; ++d) {
    int bat = bb0 + d + 8 * half;
    out[((size_t)bat * G + g) * (size_t)L2 + q] = acc[d];
  }
}

// ---------------------------------------------------------------------------
extern "C" void kernel_launch(void* const* d_in, const int* in_sizes, int n_in,
                              void* d_out, int out_size, void* d_ws, size_t ws_size,
                              hipStream_t stream) {
  (void)in_sizes; (void)n_in; (void)out_size; (void)ws_size;
  const int B = 128, N = 32768;

  const float* x = (const float*)d_in[0];
  const float* bias[5] = { (const float*)d_in[6], (const float*)d_in[7],
                           (const float*)d_in[8], (const float*)d_in[9],
                           (const float*)d_in[10] };

  // Ping-pong: stage 0,2,4,... -> ws; 1,3,5,... -> d_out (stage 9 = final -> d_out)
  float* bufs[2] = { (float*)d_ws, (float*)d_out };
  const float* src = x;
  int stage = 0;

  for (int i = 0; i < 5; ++i) {                 // analysis, levels 1..5
    int Gin  = 1 << i;
    int Lin  = N >> i;
    int Lout = Lin >> 1;
    float* dst = bufs[stage & 1];
    dim3 grid(Lout / 128, B / 16, Gin);
    wpt_analysis_wmma<<<grid, 256, 0, stream>>>(src, dst, bias[i], Gin, Lin);
    src = dst; ++stage;
  }
  for (int i = 4; i >= 0; --i) {                // synthesis, levels 5..1
    int G = 1 << i;                             // output groups
    int M = N >> (i + 1);                       // input length per channel
    float* dst = bufs[stage & 1];
    dim3 grid((2 * M) / 128, B / 16, G);
    wpt_synthesis_wmma<<<grid, 256, 0, stream>>>(src, dst, G, M);
    src = dst; ++stage;
  }
}